// GINNet_78039555768841
// MI455X (gfx1250) — compile-verified
//
#include <hip/hip_runtime.h>
#include <math.h>

#define NNODES 100000
#define NEDGES 1600000
#define NGRAPHS 128
#define DIM 64
#define NE (NNODES * DIM)

typedef __attribute__((ext_vector_type(2))) float v2f;
typedef __attribute__((ext_vector_type(8))) float v8f;

// ---------------- utility ----------------
__global__ void zero_f32(float* __restrict__ p, int n) {
  int i = blockIdx.x * blockDim.x + threadIdx.x;
  if (i < n) p[i] = 0.f;
}

// agg = (1 + eps) * h   (GIN self term folded into the scatter target)
__global__ void init_agg(const float* __restrict__ h, const float* __restrict__ epsPtr,
                         float* __restrict__ agg, int n) {
  int i = blockIdx.x * blockDim.x + threadIdx.x;
  if (i < n) agg[i] = (1.f + epsPtr[0]) * h[i];
}

// agg[dst] += h[src] over COO edges; one thread per (edge, feature).
// 25.6MB node buffer is L2-resident on MI455X (192MB L2) -> atomics run at L2 rate.
__global__ void scatter_add(const float* __restrict__ h, const int* __restrict__ ei,
                            float* __restrict__ agg) {
  int i = blockIdx.x * blockDim.x + threadIdx.x;
  if (i >= NEDGES * DIM) return;
  int e = i >> 6;
  int f = i & 63;
  int s = ei[e];
  int d = ei[NEDGES + e];
  atomicAdd(&agg[d * DIM + f], h[s * DIM + f]);
}

// ---------------- fp32 WMMA GEMM: Z[M x 64] = X[M x 64] @ W[64 x 64] ----------------
// Wave-level 16x16 tiles via V_WMMA_F32_16X16X4_F32, K walked in 16 steps of 4.
// A frag (16x4):  lane l holds M = l%16, K-half = 2*(l/16); vgpr0->K, vgpr1->K+1.
// B frag (4x16):  lane l holds N = l%16, K-half = 2*(l/16); vgpr0->K, vgpr1->K+1.
// C/D (16x16):    vgpr r, lanes 0-15 -> M=r, lanes 16-31 -> M=r+8, N = l%16.
__global__ void gemm_n64(const float* __restrict__ X, const float* __restrict__ W,
                         float* __restrict__ Z, int numRowTiles) {
  const int lane  = threadIdx.x & 31;
  const int wave  = threadIdx.x >> 5;   // 0..7
  const int ctile = wave & 3;           // which 16-col tile of N=64
  const int wrow  = wave >> 2;          // 0..1
  const int col   = ctile * 16 + (lane & 15);
  const int khalf = (lane >> 4) * 2;    // 0 or 2
  const int m     = lane & 15;

  // Preload all B fragments for this column tile (W is 16KB, L2/WGP$ resident)
  v2f bfrag[16];
#pragma unroll
  for (int kk = 0; kk < 16; ++kk) {
    int k = kk * 4 + khalf;
    bfrag[kk].x = W[k * 64 + col];
    bfrag[kk].y = W[(k + 1) * 64 + col];
  }

  for (int i = 0; i < 4; ++i) {
    int rt = blockIdx.x * 8 + wrow * 4 + i;      // wave-uniform -> EXEC stays full
    if (rt >= numRowTiles) return;
    const float* xrow = X + (size_t)(rt * 16 + m) * 64 + khalf;
    v8f acc = {0.f, 0.f, 0.f, 0.f, 0.f, 0.f, 0.f, 0.f};
#pragma unroll
    for (int kk = 0; kk < 16; ++kk) {
      v2f a;
      a.x = xrow[kk * 4];
      a.y = xrow[kk * 4 + 1];
      acc = __builtin_amdgcn_wmma_f32_16x16x4_f32(
          /*neg_a=*/false, a, /*neg_b=*/false, bfrag[kk],
          /*c_mod=*/(short)0, acc, /*reuse_a=*/false, /*reuse_b=*/false);
    }
    float* zp = Z + (size_t)rt * 16 * 64;
#pragma unroll
    for (int r = 0; r < 8; ++r) {
      int row = r + ((lane >> 4) << 3);
      zp[row * 64 + col] = acc[r];
    }
  }
}

// ---------------- BatchNorm (train-mode, biased var over all nodes) ----------------
// stats[0..63] = sum_f, stats[64..127] = sumsq_f
__global__ void bn_stats(const float* __restrict__ z, float* __restrict__ stats, int nrows) {
  const int f = threadIdx.x & 63;
  const int g = threadIdx.x >> 6;  // 0..3
  float s = 0.f, ss = 0.f;
  int base = blockIdx.x * 128;
  for (int r = g; r < 128; r += 4) {
    int row = base + r;
    if (row < nrows) {
      float v = z[(size_t)row * 64 + f];
      s += v;
      ss += v * v;
    }
  }
  __shared__ float sh[512];
  sh[threadIdx.x] = s;
  sh[256 + threadIdx.x] = ss;
  __syncthreads();
  if (g == 0) {
    for (int gg = 1; gg < 4; ++gg) {
      s += sh[gg * 64 + f];
      ss += sh[256 + gg * 64 + f];
    }
    atomicAdd(&stats[f], s);
    atomicAdd(&stats[64 + f], ss);
  }
}

// sc[0..63] = gamma*rsqrt(var+eps), sc[64..127] = beta - mean*scale
__global__ void bn_finalize(const float* __restrict__ stats, const float* __restrict__ gamma,
                            const float* __restrict__ beta, float* __restrict__ sc, int nrows) {
  int f = threadIdx.x;
  float inv_n = 1.f / (float)nrows;
  float mean = stats[f] * inv_n;
  float var = fmaxf(stats[64 + f] * inv_n - mean * mean, 0.f);
  float scale = gamma[f] * rsqrtf(var + 1e-5f);
  sc[f] = scale;
  sc[64 + f] = beta[f] - mean * scale;
}

__global__ void bn_apply_relu(const float* __restrict__ z, const float* __restrict__ sc,
                              float* __restrict__ out, int n) {
  int i = blockIdx.x * blockDim.x + threadIdx.x;
  if (i < n) {
    int f = i & 63;
    out[i] = fmaxf(fmaf(z[i], sc[f], sc[64 + f]), 0.f);
  }
}

// ---------------- global mean pool + MLP head ----------------
__global__ void pool_scatter(const float* __restrict__ emb, const int* __restrict__ batch,
                             float* __restrict__ gsum, float* __restrict__ gcnt) {
  int i = blockIdx.x * blockDim.x + threadIdx.x;
  if (i >= NE) return;
  int n = i >> 6, f = i & 63;
  int b = batch[n];
  atomicAdd(&gsum[b * 64 + f], emb[i]);
  if (f == 0) atomicAdd(&gcnt[b], 1.f);
}

// one block per graph: graph_emb -> dense(64)+ELU -> dense(2); also writes graph_emb
__global__ void head(const float* __restrict__ gsum, const float* __restrict__ gcnt,
                     const float* __restrict__ W0, const float* __restrict__ b0,
                     const float* __restrict__ W1, const float* __restrict__ b1,
                     float* __restrict__ out) {
  int g = blockIdx.x, f = threadIdx.x;
  __shared__ float se[64], shid[64];
  float cnt = fmaxf(gcnt[g], 1.f);
  float ge = gsum[g * 64 + f] / cnt;
  out[256 + g * 64 + f] = ge;  // graph_emb (output #2)
  se[f] = ge;
  __syncthreads();
  float a = b0[f];
  for (int k = 0; k < 64; ++k) a = fmaf(se[k], W0[k * 64 + f], a);
  shid[f] = a > 0.f ? a : expm1f(a);  // ELU
  __syncthreads();
  if (f < 2) {
    float o = b1[f];
    for (int k = 0; k < 64; ++k) o = fmaf(shid[k], W1[k * 2 + f], o);
    out[g * 2 + f] = o;  // logits (output #1)
  }
}

extern "C" void kernel_launch(void* const* d_in, const int* in_sizes, int n_in,
                              void* d_out, int out_size, void* d_ws, size_t ws_size,
                              hipStream_t stream) {
  const float* x     = (const float*)d_in[0];
  const int*   ei    = (const int*)d_in[1];  // [2, E]; JAX x64 off -> int32
  const int*   batch = (const int*)d_in[2];
  const float* mlp0W = (const float*)d_in[24];
  const float* mlp0b = (const float*)d_in[25];
  const float* mlp1W = (const float*)d_in[26];
  const float* mlp1b = (const float*)d_in[27];

  float* bufA  = (float*)d_ws;          // 6.4M floats
  float* bufB  = bufA + NE;             // 6.4M floats
  float* stats = bufB + NE;             // 128 (sum, sumsq)
  float* sc    = stats + 128;           // 128 (scale, shift)
  float* gsum  = sc + 128;              // 128*64
  float* gcnt  = gsum + NGRAPHS * DIM;  // 128 (contiguous after gsum)

  float* out      = (float*)d_out;
  float* node_emb = out + 256 + NGRAPHS * DIM;  // output #3

  const int EW_GRID   = (NE + 255) / 256;
  const int SC_GRID   = (NEDGES * DIM + 255) / 256;
  const int GEMM_GRID = (NNODES / 16 + 7) / 8;
  const int BN_GRID   = (NNODES + 127) / 128;

  auto layer = [&](const float* hin, int pbase, float* hout) {
    const float* epsP = (const float*)d_in[pbase];
    const float* W1 = (const float*)d_in[pbase + 1];
    const float* g1 = (const float*)d_in[pbase + 2];
    const float* b1 = (const float*)d_in[pbase + 3];
    const float* W2 = (const float*)d_in[pbase + 4];
    const float* g2 = (const float*)d_in[pbase + 5];
    const float* b2 = (const float*)d_in[pbase + 6];

    init_agg<<<EW_GRID, 256, 0, stream>>>(hin, epsP, bufA, NE);
    scatter_add<<<SC_GRID, 256, 0, stream>>>(hin, ei, bufA);
    gemm_n64<<<GEMM_GRID, 256, 0, stream>>>(bufA, W1, bufB, NNODES / 16);
    zero_f32<<<1, 128, 0, stream>>>(stats, 128);
    bn_stats<<<BN_GRID, 256, 0, stream>>>(bufB, stats, NNODES);
    bn_finalize<<<1, 64, 0, stream>>>(stats, g1, b1, sc, NNODES);
    bn_apply_relu<<<EW_GRID, 256, 0, stream>>>(bufB, sc, bufA, NE);
    gemm_n64<<<GEMM_GRID, 256, 0, stream>>>(bufA, W2, bufB, NNODES / 16);
    zero_f32<<<1, 128, 0, stream>>>(stats, 128);
    bn_stats<<<BN_GRID, 256, 0, stream>>>(bufB, stats, NNODES);
    bn_finalize<<<1, 64, 0, stream>>>(stats, g2, b2, sc, NNODES);
    bn_apply_relu<<<EW_GRID, 256, 0, stream>>>(bufB, sc, hout, NE);
  };

  layer(x,    3,  bufB);      // layer 0: in-place final ReLU into bufB
  layer(bufB, 10, bufB);      // layer 1
  layer(bufB, 17, node_emb);  // layer 2: final node embedding straight to d_out

  zero_f32<<<(NGRAPHS * DIM + NGRAPHS + 255) / 256, 256, 0, stream>>>(gsum, NGRAPHS * DIM + NGRAPHS);
  pool_scatter<<<EW_GRID, 256, 0, stream>>>(node_emb, batch, gsum, gcnt);
  head<<<NGRAPHS, 64, 0, stream>>>(gsum, gcnt, mlp0W, mlp0b, mlp1W, mlp1b, out);
}